// GCN_30288109372158
// MI455X (gfx1250) — compile-verified
//
#include <hip/hip_runtime.h>

#define N_NODES 100000
#define IN_DIM  64
#define HID_DIM 64
#define OUT_DIM 2

typedef __attribute__((ext_vector_type(2))) float v2f;
typedef __attribute__((ext_vector_type(8))) float v8f;

// ---------------------------------------------------------------------------
// Degree handling: deg[i] = 1 (self loop) + in-degree(i); then dinv = rsqrt.
// ---------------------------------------------------------------------------
__global__ void k_deg_init(float* __restrict__ deg) {
  int i = blockIdx.x * blockDim.x + threadIdx.x;
  if (i < N_NODES) deg[i] = 1.0f;
}

__global__ void k_deg_count(const long long* __restrict__ dst, float* __restrict__ deg,
                            long long E) {
  long long e = (long long)blockIdx.x * blockDim.x + threadIdx.x;
  if (e < E) atomicAdd(&deg[dst[e]], 1.0f);
}

__global__ void k_dinv(float* __restrict__ deg) {
  int i = blockIdx.x * blockDim.x + threadIdx.x;
  if (i < N_NODES) deg[i] = rsqrtf(fmaxf(deg[i], 1.0f));
}

// ---------------------------------------------------------------------------
// GEMM1: h1 = x @ W1   (fp32 WMMA 16x16x4). One wave -> 16 rows x 64 cols.
// W1 (16 KB) is staged once per block into LDS; B fragments come from LDS.
// A frag (16x4):  lane<16: M=lane, K=0..1; lane>=16: M=lane-16, K=2..3
// B frag (4x16):  v0 = K=0 (lanes 0-15) / K=2 (lanes 16-31); v1 = K=1 / K=3
// D (16x16):      VGPR v: M=v (lanes 0-15) / M=v+8 (lanes 16-31), N=lane&15
// ---------------------------------------------------------------------------
__global__ void __launch_bounds__(256) k_gemm1_wmma(const float* __restrict__ x,
                                                    const float* __restrict__ W,
                                                    float* __restrict__ h) {
  __shared__ float sW[IN_DIM * HID_DIM];          // 16 KB of the 320 KB WGP LDS

  // Cooperative coalesced copy of W1 into LDS (float4 granularity), then
  // barrier BEFORE any wave can exit (keeps the barrier membership clean).
  {
    const float4* Wv = (const float4*)W;
    float4*       Sv = (float4*)sW;
    #pragma unroll
    for (int j = threadIdx.x; j < (IN_DIM * HID_DIM) / 4; j += 256)
      Sv[j] = Wv[j];
  }
  __syncthreads();

  const int lane  = threadIdx.x & 31;
  const int wave  = threadIdx.x >> 5;
  const int strip = blockIdx.x * 8 + wave;        // one 16-row strip per wave
  if (strip * 16 >= N_NODES) return;              // wave-uniform: EXEC stays all-1s
  const int row0 = strip * 16;
  const int m    = lane & 15;
  const int kh   = (lane >> 4) << 1;              // 0 (low half) or 2 (high half)

  v8f acc0 = {}, acc1 = {}, acc2 = {}, acc3 = {};
  const float* xrow = x + (size_t)(row0 + m) * IN_DIM;

  for (int k = 0; k < IN_DIM; k += 4) {
    v2f a;
    a.x = xrow[k + kh];
    a.y = xrow[k + kh + 1];
    const float* w0 = sW + (k + kh) * HID_DIM;       // row K+kh of W1 (LDS)
    const float* w1 = sW + (k + kh + 1) * HID_DIM;   // row K+kh+1     (LDS)
    v2f b;
    b.x = w0[m +  0]; b.y = w1[m +  0];
    acc0 = __builtin_amdgcn_wmma_f32_16x16x4_f32(false, a, false, b, (short)0, acc0, false, false);
    b.x = w0[m + 16]; b.y = w1[m + 16];
    acc1 = __builtin_amdgcn_wmma_f32_16x16x4_f32(false, a, false, b, (short)0, acc1, false, false);
    b.x = w0[m + 32]; b.y = w1[m + 32];
    acc2 = __builtin_amdgcn_wmma_f32_16x16x4_f32(false, a, false, b, (short)0, acc2, false, false);
    b.x = w0[m + 48]; b.y = w1[m + 48];
    acc3 = __builtin_amdgcn_wmma_f32_16x16x4_f32(false, a, false, b, (short)0, acc3, false, false);
  }

  const int rbase = row0 + ((lane >> 4) << 3);    // +8 rows for upper half-wave
  float* hp = h + (size_t)rbase * HID_DIM + m;
  #pragma unroll
  for (int v = 0; v < 8; ++v) {
    hp[(size_t)v * HID_DIM +  0] = acc0[v];
    hp[(size_t)v * HID_DIM + 16] = acc1[v];
    hp[(size_t)v * HID_DIM + 32] = acc2[v];
    hp[(size_t)v * HID_DIM + 48] = acc3[v];
  }
}

// ---------------------------------------------------------------------------
// Layer-1 aggregation: self-loop init, edge scatter (L2-resident), bias+ReLU
// ---------------------------------------------------------------------------
__global__ void k_selfloop64(const float* __restrict__ h, const float* __restrict__ dinv,
                             float* __restrict__ agg) {
  int i = blockIdx.x * blockDim.x + threadIdx.x;  // over N_NODES*64
  if (i < N_NODES * 64) {
    int node = i >> 6;
    float di = dinv[node];
    agg[i] = di * di * h[i];
  }
}

__global__ void k_scatter64(const float* __restrict__ h, const float* __restrict__ dinv,
                            const long long* __restrict__ src, const long long* __restrict__ dst,
                            float* __restrict__ agg, long long E) {
  long long idx = (long long)blockIdx.x * blockDim.x + threadIdx.x;
  if (idx >= E * 64) return;
  long long e = idx >> 6;             // two waves cover one edge's 64 features
  int f = (int)(idx & 63);            // -> coalesced h-gather and agg-atomics
  long long s = src[e], d = dst[e];   // wave-uniform loads (hardware broadcast)
  float w = dinv[s] * dinv[d];
  atomicAdd(&agg[d * 64 + f], w * h[s * 64 + f]);
}

__global__ void k_bias_relu(const float* __restrict__ agg, const float* __restrict__ b1,
                            float* __restrict__ h) {
  int i = blockIdx.x * blockDim.x + threadIdx.x;
  if (i < N_NODES * 64) h[i] = fmaxf(agg[i] + b1[i & 63], 0.0f);
}

// ---------------------------------------------------------------------------
// GEMM2: h2 = relu(h1agg) @ W2  (64 -> 2, scalar; N=2 makes WMMA wasteful)
// Rows are 256B-aligned in the workspace -> float4 (global_load_b128) reads.
// ---------------------------------------------------------------------------
__global__ void k_gemm2(const float* __restrict__ h, const float* __restrict__ W2,
                        float* __restrict__ h2) {
  int i = blockIdx.x * blockDim.x + threadIdx.x;
  if (i >= N_NODES) return;
  const float4* hr = (const float4*)(h + (size_t)i * HID_DIM);
  float a0 = 0.0f, a1 = 0.0f;
  #pragma unroll
  for (int k4 = 0; k4 < HID_DIM / 4; ++k4) {
    float4 v = hr[k4];
    int k = k4 * 4;
    a0 += v.x * W2[(k + 0) * 2] + v.y * W2[(k + 1) * 2] +
          v.z * W2[(k + 2) * 2] + v.w * W2[(k + 3) * 2];
    a1 += v.x * W2[(k + 0) * 2 + 1] + v.y * W2[(k + 1) * 2 + 1] +
          v.z * W2[(k + 2) * 2 + 1] + v.w * W2[(k + 3) * 2 + 1];
  }
  h2[i * 2 + 0] = a0;
  h2[i * 2 + 1] = a1;
}

// ---------------------------------------------------------------------------
// Layer-2 aggregation (accumulate directly into d_out) + log_softmax in place
// ---------------------------------------------------------------------------
__global__ void k_selfloop2(const float* __restrict__ h2, const float* __restrict__ dinv,
                            float* __restrict__ out) {
  int i = blockIdx.x * blockDim.x + threadIdx.x;  // over N_NODES*2
  if (i < N_NODES * 2) {
    int node = i >> 1;
    float di = dinv[node];
    out[i] = di * di * h2[i];
  }
}

__global__ void k_scatter2(const float* __restrict__ h2, const float* __restrict__ dinv,
                           const long long* __restrict__ src, const long long* __restrict__ dst,
                           float* __restrict__ out, long long E) {
  long long idx = (long long)blockIdx.x * blockDim.x + threadIdx.x;
  if (idx >= E * 2) return;
  long long e = idx >> 1;
  int c = (int)(idx & 1);
  long long s = src[e], d = dst[e];
  float w = dinv[s] * dinv[d];
  atomicAdd(&out[d * 2 + c], w * h2[s * 2 + c]);
}

__global__ void k_logsoftmax(float* __restrict__ out, const float* __restrict__ b2) {
  int i = blockIdx.x * blockDim.x + threadIdx.x;
  if (i >= N_NODES) return;
  float a0 = out[i * 2 + 0] + b2[0];
  float a1 = out[i * 2 + 1] + b2[1];
  float mx = fmaxf(a0, a1);
  float lse = mx + logf(expf(a0 - mx) + expf(a1 - mx));
  out[i * 2 + 0] = a0 - lse;
  out[i * 2 + 1] = a1 - lse;
}

// ---------------------------------------------------------------------------
extern "C" void kernel_launch(void* const* d_in, const int* in_sizes, int n_in,
                              void* d_out, int out_size, void* d_ws, size_t ws_size,
                              hipStream_t stream) {
  const float*     x    = (const float*)d_in[0];
  const long long* ei   = (const long long*)d_in[1];   // int64 edge_index [2,E]
  const float*     W1   = (const float*)d_in[2];
  const float*     b1   = (const float*)d_in[3];
  const float*     W2   = (const float*)d_in[4];
  const float*     b2   = (const float*)d_in[5];
  float*           out  = (float*)d_out;

  const long long E   = in_sizes[1] / 2;
  const long long* src = ei;
  const long long* dst = ei + E;

  // Workspace layout (floats)
  float* ws   = (float*)d_ws;
  float* dinv = ws;                        // [0, 100000)
  float* h1   = ws + (1 << 17);            // 131072: N*64 (also holds relu output)
  float* agg1 = h1 + (size_t)N_NODES * HID_DIM;
  float* h2   = agg1 + (size_t)N_NODES * HID_DIM;

  const int B = 256;
  // degrees -> dinv
  k_deg_init  <<<(N_NODES + B - 1) / B, B, 0, stream>>>(dinv);
  k_deg_count <<<(int)((E + B - 1) / B), B, 0, stream>>>(dst, dinv, E);
  k_dinv      <<<(N_NODES + B - 1) / B, B, 0, stream>>>(dinv);
  // layer 1
  k_gemm1_wmma<<<(N_NODES / 16 + 7) / 8, B, 0, stream>>>(x, W1, h1);
  k_selfloop64<<<(N_NODES * 64 + B - 1) / B, B, 0, stream>>>(h1, dinv, agg1);
  k_scatter64 <<<(int)((E * 64 + B - 1) / B), B, 0, stream>>>(h1, dinv, src, dst, agg1, E);
  k_bias_relu <<<(N_NODES * 64 + B - 1) / B, B, 0, stream>>>(agg1, b1, h1);
  // layer 2
  k_gemm2     <<<(N_NODES + B - 1) / B, B, 0, stream>>>(h1, W2, h2);
  k_selfloop2 <<<(N_NODES * 2 + B - 1) / B, B, 0, stream>>>(h2, dinv, out);
  k_scatter2  <<<(int)((E * 2 + B - 1) / B), B, 0, stream>>>(h2, dinv, src, dst, out, E);
  k_logsoftmax<<<(N_NODES + B - 1) / B, B, 0, stream>>>(out, b2);
}